// JointPseudoLoss2_40638980555453
// MI455X (gfx1250) — compile-verified
//
#include <hip/hip_runtime.h>
#include <hip/hip_bf16.h>

// ---------------------------------------------------------------------------
// JointPseudoLoss2 for MI455X (gfx1250, wave32)
//   preds   [128,17,64,64] f32
//   targets [2,128,17,64,64] f32
//   sampleWeight [128,1] f32
// Outputs (22 f32): combined_loss, num_pseudo, num_selected,
//                   joint_score_mean[17], thr1, thr2
// ---------------------------------------------------------------------------

#define BS     128
#define K      17
#define HW     4096
#define TOT    (BS * K)      // 2176
#define SPLIT  4
#define CHUNK  (HW / SPLIT)  // 1024
#define T1     256
#define SORTN  4096
#define SELIDX 1088          // int(2176 * (1 - 0.5))
#define FLTMAX 3.402823466e38f

typedef __attribute__((ext_vector_type(2))) float v2f;
typedef __attribute__((ext_vector_type(8))) float v8f;

__device__ __forceinline__ float wredSum(float v) {
  #pragma unroll
  for (int o = 16; o > 0; o >>= 1) v += __shfl_down(v, o, 32);
  return v;
}
__device__ __forceinline__ float wredMax(float v) {
  #pragma unroll
  for (int o = 16; o > 0; o >>= 1) v = fmaxf(v, __shfl_down(v, o, 32));
  return v;
}

// ---------------------------------------------------------------------------
// Pass 1: streaming pass over preds/targets.  One block per (sample, split).
// Produces per-(b,j,split): raw sum-of-squares, max-softmax scores (v1, v2).
// ws layout (floats): [0,SPLIT*TOT)      loss partials   (split-major)
//                     [S*T, 2*S*T)       score1 partials
//                     [2*S*T, 3*S*T)     score2 partials
// ---------------------------------------------------------------------------
__global__ void __launch_bounds__(T1)
jp2_pass1(const float* __restrict__ preds,
          const float* __restrict__ targets,
          float* __restrict__ ws) {
  const int blk = blockIdx.x;          // 0 .. BS*SPLIT-1
  const int b   = blk / SPLIT;
  const int s   = blk % SPLIT;
  const int tid = threadIdx.x;

  const float* __restrict__ pB  = preds   + (size_t)b * K * HW;
  const float* __restrict__ t0B = targets + (size_t)b * K * HW;
  const float* __restrict__ t1B = targets + (size_t)(BS + b) * K * HW;

  float lsum[K], s1[K], s2[K];
  #pragma unroll
  for (int j = 0; j < K; ++j) { lsum[j] = 0.f; s1[j] = 0.f; s2[j] = 0.f; }

  const int iBeg = s * CHUNK;
  for (int i = iBeg + tid; i < iBeg + CHUNK; i += T1) {
    float v1v[K], v2v[K], e[K];
    #pragma unroll
    for (int j = 0; j < K; ++j) v1v[j] = pB[j * HW + i];
    #pragma unroll
    for (int j = 0; j < K; ++j)
      v2v[j] = 0.5f * (t0B[j * HW + i] + t1B[j * HW + i]);

    // per-joint squared error accumulation
    #pragma unroll
    for (int j = 0; j < K; ++j) {
      float d = v1v[j] - v2v[j];
      lsum[j] = fmaf(d, d, lsum[j]);
    }
    // softmax over joints (v1), running max over spatial
    {
      float m = v1v[0];
      #pragma unroll
      for (int j = 1; j < K; ++j) m = fmaxf(m, v1v[j]);
      float sum = 0.f;
      #pragma unroll
      for (int j = 0; j < K; ++j) { e[j] = __expf(v1v[j] - m); sum += e[j]; }
      float inv = 1.0f / sum;
      #pragma unroll
      for (int j = 0; j < K; ++j) s1[j] = fmaxf(s1[j], e[j] * inv);
    }
    // softmax over joints (v2), running max over spatial
    {
      float m = v2v[0];
      #pragma unroll
      for (int j = 1; j < K; ++j) m = fmaxf(m, v2v[j]);
      float sum = 0.f;
      #pragma unroll
      for (int j = 0; j < K; ++j) { e[j] = __expf(v2v[j] - m); sum += e[j]; }
      float inv = 1.0f / sum;
      #pragma unroll
      for (int j = 0; j < K; ++j) s2[j] = fmaxf(s2[j], e[j] * inv);
    }
  }

  // wave32 reduction, then across the 8 waves through LDS
  #pragma unroll
  for (int j = 0; j < K; ++j) {
    lsum[j] = wredSum(lsum[j]);
    s1[j]   = wredMax(s1[j]);
    s2[j]   = wredMax(s2[j]);
  }
  __shared__ float shL[T1 / 32][K], shA[T1 / 32][K], shB[T1 / 32][K];
  const int wave = tid >> 5, lane = tid & 31;
  if (lane == 0) {
    #pragma unroll
    for (int j = 0; j < K; ++j) { shL[wave][j] = lsum[j]; shA[wave][j] = s1[j]; shB[wave][j] = s2[j]; }
  }
  __syncthreads();
  if (tid < K) {
    float L = 0.f, a = 0.f, c = 0.f;
    #pragma unroll
    for (int w = 0; w < T1 / 32; ++w) {
      L += shL[w][tid];
      a = fmaxf(a, shA[w][tid]);
      c = fmaxf(c, shB[w][tid]);
    }
    const int idx = b * K + tid;
    ws[             s * TOT + idx] = L;
    ws[    SPLIT * TOT + s * TOT + idx] = a;
    ws[2 * SPLIT * TOT + s * TOT + idx] = c;
  }
}

// ---------------------------------------------------------------------------
// Pass 2: single workgroup finalize.
// ---------------------------------------------------------------------------
__device__ float bitonic_thr(const float* __restrict__ src, float* buf, int tid) {
  for (int i = tid; i < SORTN; i += T1) buf[i] = (i < TOT) ? src[i] : FLTMAX;
  __syncthreads();
  for (int k = 2; k <= SORTN; k <<= 1) {
    for (int j = k >> 1; j > 0; j >>= 1) {
      for (int i = tid; i < SORTN; i += T1) {
        int ixj = i ^ j;
        if (ixj > i) {
          float a = buf[i], c = buf[ixj];
          bool up = ((i & k) == 0);
          if ((a > c) == up) { buf[i] = c; buf[ixj] = a; }
        }
      }
      __syncthreads();
    }
  }
  float t = buf[SELIDX];   // ascending sort -> exact flat[1088]
  __syncthreads();
  return t;
}

__global__ void __launch_bounds__(T1)
jp2_pass2(const float* __restrict__ ws,
          const float* __restrict__ sw,
          float* __restrict__ out) {
  __shared__ float sc1[TOT], sc2[TOT], lossF[TOT];
  __shared__ float buf[SORTN];
  __shared__ float whS[BS];
  __shared__ float sCnt;
  __shared__ float redBuf[3][T1 / 32];

  const int tid = threadIdx.x;

  // combine split partials; fold mean (1/HW) and sampleWeight into loss
  for (int idx = tid; idx < TOT; idx += T1) {
    float L = 0.f, a = 0.f, c = 0.f;
    #pragma unroll
    for (int s = 0; s < SPLIT; ++s) {
      L += ws[             s * TOT + idx];
      a = fmaxf(a, ws[    SPLIT * TOT + s * TOT + idx]);
      c = fmaxf(c, ws[2 * SPLIT * TOT + s * TOT + idx]);
    }
    const int b = idx / K;
    lossF[idx] = L * (1.0f / (float)HW) * sw[b];
    sc1[idx] = a;
    sc2[idx] = c;
  }
  if (tid == 0) {
    float cnt = 0.f;
    for (int b = 0; b < BS; ++b) cnt += (sw[b] > 0.f) ? 1.f : 0.f;
    sCnt = cnt;
  }
  __syncthreads();

  // weight vector for joint means: indicator(sw>0) * 0.5 / cnt
  if (tid < BS) whS[tid] = (sw[tid] > 0.f) ? (0.5f / sCnt) : 0.f;
  __syncthreads();

  // exact (1-selRate) quantiles via in-LDS bitonic sort
  const float thr1 = bitonic_thr(sc1, buf, tid);
  const float thr2 = bitonic_thr(sc2, buf, tid);

  // masked sums / counts
  float cl = 0.f, np = 0.f, ns = 0.f;
  for (int idx = tid; idx < TOT; idx += T1) {
    float lf  = lossF[idx];
    float msk = ((sc1[idx] >= thr1) && (sc2[idx] >= thr2)) ? 1.f : 0.f;
    cl += lf * msk;
    np += (lf > 0.f) ? 1.f : 0.f;
    ns += msk;
  }
  cl = wredSum(cl); np = wredSum(np); ns = wredSum(ns);
  const int wave = tid >> 5, lane = tid & 31;
  if (lane == 0) { redBuf[0][wave] = cl; redBuf[1][wave] = np; redBuf[2][wave] = ns; }
  __syncthreads();
  if (tid == 0) {
    float c = 0.f, n = 0.f, s = 0.f;
    #pragma unroll
    for (int w = 0; w < T1 / 32; ++w) { c += redBuf[0][w]; n += redBuf[1][w]; s += redBuf[2][w]; }
    out[0] = c;            // combined_loss
    out[1] = n;            // num_pseudo (as float)
    out[2] = s;            // num_selected (as float)
    out[20] = thr1;        // rateThr1
    out[21] = thr2;        // rateThr2
    // joint 16 (the k=17 leftover row) scalar path
    float j16 = 0.f;
    for (int b = 0; b < BS; ++b)
      j16 += (sc1[b * K + 16] + sc2[b * K + 16]) * whS[b];
    out[3 + 16] = j16;
  }
  __syncthreads();

  // joints 0..15: 16x128 score matrix times weight vector via
  // V_WMMA_F32_16X16X4_F32 chain (32 K-chunks of 4 batches each).
  // A layout (16x4 f32): lanes 0-15 hold {A[m,0],A[m,1]}, lanes 16-31 hold
  // {A[m,2],A[m,3]}.  B (4x16) broadcasts w[k] across all columns, row-striped
  // {rows k, k+2} per VGPR.  All 8 waves compute identically (EXEC all ones);
  // wave 0 extracts column 0 of D.
  {
    const int m  = tid & 15;
    const int hi = (tid >> 4) & 1;
    v8f acc = {0.f, 0.f, 0.f, 0.f, 0.f, 0.f, 0.f, 0.f};
    for (int c = 0; c < 32; ++c) {
      const int b0 = 4 * c + 2 * hi;
      v2f av, bv;
      av.x = sc1[b0 * K + m]       + sc2[b0 * K + m];
      av.y = sc1[(b0 + 1) * K + m] + sc2[(b0 + 1) * K + m];
      bv.x = whS[b0];
      bv.y = whS[b0 + 1];
      acc = __builtin_amdgcn_wmma_f32_16x16x4_f32(
          /*neg_a=*/false, av, /*neg_b=*/false, bv,
          /*c_mod=*/(short)0, acc, /*reuse_a=*/false, /*reuse_b=*/false);
    }
    // D layout: VGPR r -> rows r (lanes 0-15) and r+8 (lanes 16-31); all
    // columns equal, take N=0 (lane 0) and N=0 of upper half (lane 16).
    if (tid == 0) {
      #pragma unroll
      for (int r = 0; r < 8; ++r) out[3 + r] = acc[r];
    }
    if (tid == 16) {
      #pragma unroll
      for (int r = 0; r < 8; ++r) out[3 + 8 + r] = acc[r];
    }
  }
}

// ---------------------------------------------------------------------------
extern "C" void kernel_launch(void* const* d_in, const int* in_sizes, int n_in,
                              void* d_out, int out_size, void* d_ws, size_t ws_size,
                              hipStream_t stream) {
  const float* preds   = (const float*)d_in[0];
  const float* targets = (const float*)d_in[1];
  const float* sw      = (const float*)d_in[2];
  float* out = (float*)d_out;
  float* ws  = (float*)d_ws;   // needs 3*SPLIT*TOT floats = 104,448 bytes

  jp2_pass1<<<BS * SPLIT, T1, 0, stream>>>(preds, targets, ws);
  jp2_pass2<<<1, T1, 0, stream>>>(ws, sw, out);
}